// Aggregator_36438502539669
// MI455X (gfx1250) — compile-verified
//
#include <hip/hip_runtime.h>
#include <hip/hip_bf16.h>

#define N_USERS    100000
#define N_ENTITIES 200000
#define N_EDGES_C  500000
#define DIM        64
#define NF         3

typedef float v2f __attribute__((ext_vector_type(2)));
typedef float v8f __attribute__((ext_vector_type(8)));

#define XS_STRIDE 132                 // 128 + 4 pad (banks spread by 4*row)
#define RS_STRIDE 68                  // 64 + 4 pad
#define OFF_XS   0
#define OFF_RS   (16 * XS_STRIDE)     // 2112
#define OFF_UIDX (OFF_RS + 16 * RS_STRIDE)  // 3200
#define OFF_IIDX (OFF_UIDX + 16)
#define OFF_T    (OFF_IIDX + 16)
#define OFF_SIM  (OFF_T + 16)
#define WAVE_F   (OFF_SIM + 16)       // 3264 floats per wave region (13056 B)
#define WPB      4                    // waves per block for edge kernels

__device__ __forceinline__ float lrelu(float x) { return x > 0.0f ? x : 0.01f * x; }

// One wave computes rel[16x64] = leaky_relu(X[16x128] @ W^T + b) into Rs.
// X staged in LDS (cols 0-63 = user row, 64-127 = item row).
// swapHalves: logically use concat(item,user) (column index XOR 64), realized
// by swapping the base pointers of the two 16-step K sub-loops.
__device__ __forceinline__ void wave_gemm_relu(const float* __restrict__ W,
                                               const float* __restrict__ bvec,
                                               const float* Xs, float* Rs,
                                               int lane, bool swapHalves) {
  const int c = lane & 15;        // row M for A-frag, col N for B-frag
  const int h = lane >> 4;        // lane-half -> K sub-pair (+0 or +2)
  const float* xbase = Xs + c * XS_STRIDE + 2 * h;
  const float* xlo = swapHalves ? (xbase + 64) : xbase;   // K 0..63 source
  const float* xhi = swapHalves ? xbase : (xbase + 64);   // K 64..127 source
  for (int nt = 0; nt < 4; ++nt) {
    const int n = nt * 16 + c;
    const float bias = bvec[n];
    const float* wbase = W + n * 128 + 2 * h;
    v8f acc = {};
#pragma unroll
    for (int k = 0; k < 16; ++k) {        // constant 16B strides -> imm offsets
      float2 xv = *reinterpret_cast<const float2*>(xlo + 4 * k);
      float2 wv = *reinterpret_cast<const float2*>(wbase + 4 * k);
      v2f a; a[0] = xv.x; a[1] = xv.y;
      v2f b; b[0] = wv.x; b[1] = wv.y;
      acc = __builtin_amdgcn_wmma_f32_16x16x4_f32(false, a, false, b,
                                                  (short)0, acc, false, false);
    }
#pragma unroll
    for (int k = 0; k < 16; ++k) {
      float2 xv = *reinterpret_cast<const float2*>(xhi + 4 * k);
      float2 wv = *reinterpret_cast<const float2*>(wbase + 64 + 4 * k);
      v2f a; a[0] = xv.x; a[1] = xv.y;
      v2f b; b[0] = wv.x; b[1] = wv.y;
      acc = __builtin_amdgcn_wmma_f32_16x16x4_f32(false, a, false, b,
                                                  (short)0, acc, false, false);
    }
#pragma unroll
    for (int r = 0; r < 8; ++r) {
      Rs[(r + 8 * h) * RS_STRIDE + n] = lrelu(acc[r] + bias);
    }
  }
}

// Stage 16 gathered rows of concat(user_emb[u], entity_emb[i]) into LDS using
// CDNA5 async global->LDS copies (ASYNCcnt), then drain the counter.
__device__ __forceinline__ void stage_tile_async(const float* __restrict__ user_emb,
                                                 const float* __restrict__ entity_emb,
                                                 const int* uS, const int* iS,
                                                 float* Xs, int lane) {
  for (int it = 0; it < 16; ++it) {
    int cidx = it * 32 + lane;
    int m = cidx >> 5;
    int q = cidx & 31;
    const float* src;
    int dst;
    if (q < 16) { src = user_emb + (size_t)uS[m] * DIM + q * 4;          dst = m * XS_STRIDE + q * 4; }
    else        { src = entity_emb + (size_t)iS[m] * DIM + (q - 16) * 4; dst = m * XS_STRIDE + 64 + (q - 16) * 4; }
    // Low 32 bits of the generic shared-pointer == wave-relative LDS offset.
    unsigned int ldsoff = (unsigned int)(size_t)(&Xs[dst]);
    asm volatile("global_load_async_to_lds_b128 %0, %1, off"
                 :: "v"(ldsoff), "v"(src)
                 : "memory");
  }
  asm volatile("s_wait_asynccnt 0" ::: "memory");
}

__global__ void zero_kernel(float4* p, size_t n4) {
  size_t i = (size_t)blockIdx.x * blockDim.x + threadIdx.x;
  size_t stride = (size_t)gridDim.x * blockDim.x;
  float4 z = make_float4(0.f, 0.f, 0.f, 0.f);
  for (; i < n4; i += stride) p[i] = z;
}

// Pass A: relation_ui GEMM -> argmax rel_type; first segment sums.
__global__ void __launch_bounds__(32 * WPB)
pass_a(const float* __restrict__ entity_emb, const float* __restrict__ user_emb,
       const float* __restrict__ latent, const int* __restrict__ user_index,
       const int* __restrict__ item_index, const float* __restrict__ WW_w,
       const float* __restrict__ WW_b, int* __restrict__ rel_type,
       float* __restrict__ cnt, float* __restrict__ cnt_i,
       float* __restrict__ usum, float* __restrict__ ysum, int ntiles) {
  __shared__ __align__(16) float smem[WPB * WAVE_F];
  const int lane = threadIdx.x & 31;
  const int wid = threadIdx.x >> 5;
  float* wsm = smem + wid * WAVE_F;
  float* Xs = wsm + OFF_XS;
  float* Rs = wsm + OFF_RS;
  int* uS = (int*)(wsm + OFF_UIDX);
  int* iS = (int*)(wsm + OFF_IIDX);
  int* tS = (int*)(wsm + OFF_T);

  const int tile = blockIdx.x * WPB + wid;
  if (tile >= ntiles) return;
  const int e0 = tile * 16;
  const int ecnt = min(16, N_EDGES_C - e0);

  if (lane < 16) {
    int e = e0 + min(lane, ecnt - 1);
    uS[lane] = user_index[e];
    iS[lane] = item_index[e];
  }
  stage_tile_async(user_emb, entity_emb, uS, iS, Xs, lane);
  wave_gemm_relu(WW_w, WW_b, Xs, Rs, lane, false);

  if (lane < 16) {
    const int m = lane;
    float best = -3.4e38f; int bi = 0;
    for (int f = 0; f < NF; ++f) {
      float s = 0.f;
      const float* lrow = latent + f * DIM;
      for (int d = 0; d < DIM; ++d) s += Rs[m * RS_STRIDE + d] * lrow[d];
      if (s > best) { best = s; bi = f; }   // strict > == first-max argmax
    }
    tS[m] = bi;
    if (m < ecnt) {
      rel_type[e0 + m] = bi;
      unsafeAtomicAdd(&cnt[(size_t)bi * N_USERS + uS[m]], 1.0f);
      unsafeAtomicAdd(&cnt_i[iS[m]], 1.0f);
    }
  }
  for (int m = 0; m < ecnt; ++m) {
    const int t = tS[m], uu = uS[m], ii = iS[m];
    float* ub = usum + ((size_t)t * N_USERS + uu) * DIM;
    unsafeAtomicAdd(ub + lane,      Xs[m * XS_STRIDE + 64 + lane]);
    unsafeAtomicAdd(ub + lane + 32, Xs[m * XS_STRIDE + 64 + lane + 32]);
    float* yb = ysum + (size_t)ii * DIM;
    unsafeAtomicAdd(yb + lane,      Xs[m * XS_STRIDE + lane]);
    unsafeAtomicAdd(yb + lane + 32, Xs[m * XS_STRIDE + lane + 32]);
  }
}

// Pass C: in-place divide rows by max(count,1).
__global__ void pass_c(float* __restrict__ buf, const float* __restrict__ cnt, size_t total) {
  size_t i = (size_t)blockIdx.x * blockDim.x + threadIdx.x;
  if (i >= total) return;
  float dn = cnt[i >> 6];
  dn = dn > 1.0f ? dn : 1.0f;
  buf[i] = buf[i] / dn;
}

// Pass D: recompute both relation GEMMs, sim dots, second segment sums.
__global__ void __launch_bounds__(32 * WPB)
pass_d(const float* __restrict__ entity_emb, const float* __restrict__ user_emb,
       const int* __restrict__ user_index, const int* __restrict__ item_index,
       const float* __restrict__ WW_w, const float* __restrict__ WW_b,
       const float* __restrict__ WWi_w, const float* __restrict__ WWi_b,
       const int* __restrict__ rel_type, const float* __restrict__ u_,
       const float* __restrict__ y_, float* __restrict__ usum2,
       float* __restrict__ ysum2, int ntiles) {
  __shared__ __align__(16) float smem[WPB * WAVE_F];
  const int lane = threadIdx.x & 31;
  const int wid = threadIdx.x >> 5;
  float* wsm = smem + wid * WAVE_F;
  float* Xs = wsm + OFF_XS;
  float* Rs = wsm + OFF_RS;
  int* uS = (int*)(wsm + OFF_UIDX);
  int* iS = (int*)(wsm + OFF_IIDX);
  int* tS = (int*)(wsm + OFF_T);
  float* simS = wsm + OFF_SIM;

  const int tile = blockIdx.x * WPB + wid;
  if (tile >= ntiles) return;
  const int e0 = tile * 16;
  const int ecnt = min(16, N_EDGES_C - e0);

  if (lane < 16) {
    int e = e0 + min(lane, ecnt - 1);
    uS[lane] = user_index[e];
    iS[lane] = item_index[e];
    tS[lane] = rel_type[e];
  }
  stage_tile_async(user_emb, entity_emb, uS, iS, Xs, lane);

  // direction u->i
  wave_gemm_relu(WW_w, WW_b, Xs, Rs, lane, false);
  if (lane < 16) {
    const int m = lane;
    const float* ur = u_ + ((size_t)tS[m] * N_USERS + uS[m]) * DIM;
    float s = 0.f;
    for (int d = 0; d < DIM; ++d) s += Rs[m * RS_STRIDE + d] * ur[d];
    simS[m] = s;
  }
  for (int m = 0; m < ecnt; ++m) {
    const float s = simS[m];
    float* ub = usum2 + ((size_t)tS[m] * N_USERS + uS[m]) * DIM;
    unsafeAtomicAdd(ub + lane,      s * Xs[m * XS_STRIDE + 64 + lane]);
    unsafeAtomicAdd(ub + lane + 32, s * Xs[m * XS_STRIDE + 64 + lane + 32]);
  }

  // direction i<-u (concat order swapped)
  wave_gemm_relu(WWi_w, WWi_b, Xs, Rs, lane, true);
  if (lane < 16) {
    const int m = lane;
    const float* yr = y_ + (size_t)iS[m] * DIM;
    float s = 0.f;
    for (int d = 0; d < DIM; ++d) s += Rs[m * RS_STRIDE + d] * yr[d];
    simS[m] = s;
  }
  for (int m = 0; m < ecnt; ++m) {
    const float s = simS[m];
    float* yb = ysum2 + (size_t)iS[m] * DIM;
    unsafeAtomicAdd(yb + lane,      s * Xs[m * XS_STRIDE + lane]);
    unsafeAtomicAdd(yb + lane + 32, s * Xs[m * XS_STRIDE + lane + 32]);
  }
}

// Pass E: per-user squash-normalize per factor, softmax(w) blend, +user_emb.
__global__ void __launch_bounds__(256)
pass_e(const float* __restrict__ user_emb, const float* __restrict__ wvec,
       const float* __restrict__ usum2, const float* __restrict__ cnt,
       float* __restrict__ out_user) {
  const int lane = threadIdx.x & 31;
  const int wid = threadIdx.x >> 5;
  const int u = blockIdx.x * 8 + wid;
  if (u >= N_USERS) return;
  const float w0 = wvec[0], w1 = wvec[1], w2 = wvec[2];
  const float mx = fmaxf(w0, fmaxf(w1, w2));
  const float x0 = expf(w0 - mx), x1 = expf(w1 - mx), x2 = expf(w2 - mx);
  const float inv = 1.0f / (x0 + x1 + x2);
  const float wsx[3] = {x0 * inv, x1 * inv, x2 * inv};
  const float ue0 = user_emb[(size_t)u * DIM + lane];
  const float ue1 = user_emb[(size_t)u * DIM + lane + 32];
  float r0 = 0.f, r1 = 0.f;
  for (int t = 0; t < NF; ++t) {
    float dn = cnt[(size_t)t * N_USERS + u];
    dn = dn > 1.0f ? dn : 1.0f;
    const float* row = usum2 + ((size_t)t * N_USERS + u) * DIM;
    float v0 = row[lane] / dn, v1 = row[lane + 32] / dn;
    float ss = v0 * v0 + v1 * v1;
    for (int off = 16; off > 0; off >>= 1) ss += __shfl_xor(ss, off, 32);
    const float n = sqrtf(ss);
    const float scale = (ss / (ss + 1.0f)) / fmaxf(n, 1e-12f);
    r0 += wsx[t] * (scale * v0 + ue0);
    r1 += wsx[t] * (scale * v1 + ue1);
  }
  out_user[(size_t)u * DIM + lane] = r0;
  out_user[(size_t)u * DIM + lane + 32] = r1;
}

// Pass F: per-entity squash-normalize + entity_emb.
__global__ void __launch_bounds__(256)
pass_f(const float* __restrict__ entity_emb, const float* __restrict__ ysum2,
       const float* __restrict__ cnt_i, float* __restrict__ out_ent) {
  const int lane = threadIdx.x & 31;
  const int wid = threadIdx.x >> 5;
  const int i = blockIdx.x * 8 + wid;
  if (i >= N_ENTITIES) return;
  float dn = cnt_i[i];
  dn = dn > 1.0f ? dn : 1.0f;
  const float* row = ysum2 + (size_t)i * DIM;
  float v0 = row[lane] / dn, v1 = row[lane + 32] / dn;
  float ss = v0 * v0 + v1 * v1;
  for (int off = 16; off > 0; off >>= 1) ss += __shfl_xor(ss, off, 32);
  const float n = sqrtf(ss);
  const float scale = (ss / (ss + 1.0f)) / fmaxf(n, 1e-12f);
  out_ent[(size_t)i * DIM + lane] = scale * v0 + entity_emb[(size_t)i * DIM + lane];
  out_ent[(size_t)i * DIM + lane + 32] = scale * v1 + entity_emb[(size_t)i * DIM + lane + 32];
}

extern "C" void kernel_launch(void* const* d_in, const int* in_sizes, int n_in,
                              void* d_out, int out_size, void* d_ws, size_t ws_size,
                              hipStream_t stream) {
  const float* entity_emb = (const float*)d_in[0];
  const float* user_emb   = (const float*)d_in[1];
  const float* latent     = (const float*)d_in[2];
  const int*   user_index = (const int*)d_in[3];
  const int*   item_index = (const int*)d_in[4];
  const float* WW_w  = (const float*)d_in[5];
  const float* WW_b  = (const float*)d_in[6];
  const float* WWi_w = (const float*)d_in[7];
  const float* WWi_b = (const float*)d_in[8];
  const float* wvec  = (const float*)d_in[9];

  // Workspace layout (floats); all accumulation buffers contiguous for zeroing.
  float* wsf = (float*)d_ws;
  size_t o = 0;
  float* cnt   = wsf + o; o += (size_t)NF * N_USERS;
  float* cnt_i = wsf + o; o += (size_t)N_ENTITIES;
  float* usum  = wsf + o; o += (size_t)NF * N_USERS * DIM;
  float* usum2 = wsf + o; o += (size_t)NF * N_USERS * DIM;
  float* ysum  = wsf + o; o += (size_t)N_ENTITIES * DIM;
  float* ysum2 = wsf + o; o += (size_t)N_ENTITIES * DIM;
  const size_t zero_floats = o;                 // 64,500,000 (divisible by 4)
  int* rel_type = (int*)(wsf + o);

  float* out_ent  = (float*)d_out;
  float* out_user = out_ent + (size_t)N_ENTITIES * DIM;

  zero_kernel<<<4096, 256, 0, stream>>>((float4*)wsf, zero_floats / 4);

  const int ntiles = (N_EDGES_C + 15) / 16;
  const int eb = (ntiles + WPB - 1) / WPB;
  pass_a<<<eb, 32 * WPB, 0, stream>>>(entity_emb, user_emb, latent, user_index,
                                      item_index, WW_w, WW_b, rel_type, cnt,
                                      cnt_i, usum, ysum, ntiles);

  const size_t nu = (size_t)NF * N_USERS * DIM;
  const size_t ne = (size_t)N_ENTITIES * DIM;
  pass_c<<<(int)((nu + 255) / 256), 256, 0, stream>>>(usum, cnt, nu);
  pass_c<<<(int)((ne + 255) / 256), 256, 0, stream>>>(ysum, cnt_i, ne);

  pass_d<<<eb, 32 * WPB, 0, stream>>>(entity_emb, user_emb, user_index,
                                      item_index, WW_w, WW_b, WWi_w, WWi_b,
                                      rel_type, usum, ysum, usum2, ysum2, ntiles);

  pass_e<<<(N_USERS + 7) / 8, 256, 0, stream>>>(user_emb, wvec, usum2, cnt, out_user);
  pass_f<<<(N_ENTITIES + 7) / 8, 256, 0, stream>>>(entity_emb, ysum2, cnt_i, out_ent);
}